// HungarianLossDimensionMatching_84189948936344
// MI455X (gfx1250) — compile-verified
//
#include <hip/hip_runtime.h>

// loss = sum_dim sum_b (2/N) * sum_i |sort(p1)[i] - sort(p2)[i]|
// 1-D optimal transport: sorted matching == Hungarian optimum for |x - y| cost.

typedef float v2f __attribute__((ext_vector_type(2)));
typedef float v8f __attribute__((ext_vector_type(8)));

#define NPTS 256
#define NBATCH 8
#define NDIM 2
#define PROJ_EPS 1e-8f

__global__ __launch_bounds__(NPTS) void wloss_sort_kernel(
    const float* __restrict__ set1,   // [B, N, 4]
    const float* __restrict__ set2,   // [B, N, 4]
    float* __restrict__ ws) {         // [16] per-(dim,batch) W1 costs
  __shared__ float a1[NPTS];
  __shared__ float a2[NPTS];
  __shared__ float red[32];

  const int tid = threadIdx.x;
  const int bid = blockIdx.x;        // 0..15
  const int dim = bid >> 3;          // 0..1
  const int b   = bid & 7;           // 0..7

  // One float4 per point: (x, y, c2, c3)
  const float4 q1 = ((const float4*)set1)[b * NPTS + tid];
  const float4 q2 = ((const float4*)set2)[b * NPTS + tid];

  const float fd = (float)dim;

  // set1: col2 holds exact 0.0/1.0 labels -> exact equality like jnp
  const bool keep1 = (q1.z == fd);
  const float x1 = keep1 ? q1.x : 0.0f;
  const float y1 = keep1 ? q1.y : 0.0f;

  // set2: label = argmax(col2, col3); first-index tie-break => (c3 > c2) ? 1 : 0
  const int lab2 = (q2.w > q2.z) ? 1 : 0;
  const bool keep2 = (lab2 == dim);
  const float x2 = keep2 ? q2.x : 0.0f;
  const float y2 = keep2 ? q2.y : 0.0f;

  // projection: p = x + sqrt(y + eps)/2  (masked points -> 5e-5, NOT 0)
  a1[tid] = x1 + 0.5f * __builtin_sqrtf(y1 + PROJ_EPS);
  a2[tid] = x2 + 0.5f * __builtin_sqrtf(y2 + PROJ_EPS);
  __syncthreads();

  // Bitonic sort both arrays ascending (disjoint pairs -> no intra-stage race)
  for (int k = 2; k <= NPTS; k <<= 1) {
    for (int j = k >> 1; j > 0; j >>= 1) {
      const int ixj = tid ^ j;
      if (ixj > tid) {
        const bool up = ((tid & k) == 0);
        float u = a1[tid], v = a1[ixj];
        if ((u > v) == up) { a1[tid] = v; a1[ixj] = u; }
        float p = a2[tid], q = a2[ixj];
        if ((p > q) == up) { a2[tid] = q; a2[ixj] = p; }
      }
      __syncthreads();
    }
  }

  // Monotone matching cost terms
  a1[tid] = __builtin_fabsf(a1[tid] - a2[tid]);
  __syncthreads();

  // Wave 0 reduces the 256 diffs on the matrix pipe.
  // With B == all-ones and C accumulated, sum over ALL elements of D equals
  // 16 * (sum of all A elements) regardless of the A/B operand striping.
  if (tid < 32) {                    // wave-uniform: EXEC all-ones in wave 0
    v8f acc = {};
    v2f ones;
    ones.x = 1.0f;
    ones.y = 1.0f;
#pragma unroll
    for (int c = 0; c < 4; ++c) {
      v2f av;
      av.x = a1[c * 64 + tid];
      av.y = a1[c * 64 + 32 + tid];
      acc = __builtin_amdgcn_wmma_f32_16x16x4_f32(
          /*neg_a=*/false, av, /*neg_b=*/false, ones,
          /*c_mod=*/(short)0, acc, /*reuse_a=*/false, /*reuse_b=*/false);
    }
    float part = 0.0f;
#pragma unroll
    for (int r = 0; r < 8; ++r) part += acc[r];
    red[tid] = part;
  }
  __syncthreads();

  if (tid == 0) {
    float s = 0.0f;
    for (int l = 0; l < 32; ++l) s += red[l];   // fixed order: deterministic
    ws[bid] = s * (1.0f / 16.0f);               // undo the 16x replication
  }
}

__global__ void wloss_finalize(const float* __restrict__ ws,
                               float* __restrict__ out) {
  if (threadIdx.x == 0 && blockIdx.x == 0) {
    float s = 0.0f;
    for (int i = 0; i < NDIM * NBATCH; ++i) s += ws[i];  // fixed order
    out[0] = s * (2.0f / (float)NPTS);  // each batch matrix appears twice; mean over N
  }
}

extern "C" void kernel_launch(void* const* d_in, const int* in_sizes, int n_in,
                              void* d_out, int out_size, void* d_ws, size_t ws_size,
                              hipStream_t stream) {
  const float* set1 = (const float*)d_in[0];
  const float* set2 = (const float*)d_in[1];
  float* out = (float*)d_out;
  float* ws  = (float*)d_ws;   // needs 16 floats

  (void)in_sizes; (void)n_in; (void)out_size; (void)ws_size;

  wloss_sort_kernel<<<NDIM * NBATCH, NPTS, 0, stream>>>(set1, set2, ws);
  wloss_finalize<<<1, 32, 0, stream>>>(ws, out);
}